// SimpleModel_47519518163436
// MI455X (gfx1250) — compile-verified
//
#include <hip/hip_runtime.h>
#include <hip/hip_bf16.h>
#include <math.h>

#define N_NODES_C 200000
#define NUM_GRAPHS_C 512
#define CCH 128
#define BN_EPS_C 1e-5f

typedef __attribute__((ext_vector_type(2))) float v2f;
typedef __attribute__((ext_vector_type(8))) float v8f;

// ---------------- init helpers ----------------
__global__ void fill_f32(float* __restrict__ p, float v, int n) {
  int i = blockIdx.x * blockDim.x + threadIdx.x;
  if (i < n) p[i] = v;
}

// ---------------- h = x @ conv_w  (WMMA f32 16x16x4, K=4 exactly) ----------------
__global__ void __launch_bounds__(32) transform_wmma(const float* __restrict__ x,
                                                     const float* __restrict__ w,
                                                     float* __restrict__ h) {
  const int tile = blockIdx.x;          // 12500 tiles of 16 rows
  const int lane = threadIdx.x;         // wave32
  const int half = lane >> 4;
  const int m    = lane & 15;
  const int row  = tile * 16 + m;

  // A fragment 16x4: vgpr0 = K=(2*half), vgpr1 = K=(2*half+1)  -> contiguous pair
  v2f a;
  a.x = x[row * 4 + half * 2 + 0];
  a.y = x[row * 4 + half * 2 + 1];

  for (int ct = 0; ct < CCH / 16; ++ct) {
    const int col = ct * 16 + m;
    v2f b;
    b.x = w[(half * 2 + 0) * CCH + col];
    b.y = w[(half * 2 + 1) * CCH + col];
    v8f c = {};
    c = __builtin_amdgcn_wmma_f32_16x16x4_f32(false, a, false, b, (short)0, c, false, false);
    const int rbase = tile * 16 + half * 8;
#pragma unroll
    for (int r = 0; r < 8; ++r)
      h[(rbase + r) * CCH + col] = c[r];
  }
}

// ---------------- degrees (self-loop already folded in via init deg=1) ----------------
__global__ void degree_kernel(const int* __restrict__ src, const int* __restrict__ dst,
                              float* __restrict__ deg, int E) {
  int e = blockIdx.x * blockDim.x + threadIdx.x;
  if (e >= E) return;
  atomicAdd(&deg[dst[e]], 1.0f);
  atomicAdd(&deg[src[e]], 1.0f);   // reverse edge from to_undirected
}

__global__ void dinv_kernel(const float* __restrict__ deg, float* __restrict__ dinv, int n) {
  int i = blockIdx.x * blockDim.x + threadIdx.x;
  if (i < n) dinv[i] = rsqrtf(deg[i]);
}

// ---------------- edge scatter: one wave per directed edge, both directions ----------------
__global__ void __launch_bounds__(256) scatter_kernel(const int* __restrict__ src,
                                                      const int* __restrict__ dst,
                                                      const float* __restrict__ h,
                                                      const float* __restrict__ dinv,
                                                      float* __restrict__ agg, int E) {
  int e    = (int)((blockIdx.x * blockDim.x + threadIdx.x) >> 5);
  int lane = threadIdx.x & 31;
  if (e >= E) return;
  int s = src[e], d = dst[e];
  float coef = dinv[s] * dinv[d];
  float4 vs = *(const float4*)(h + (size_t)s * CCH + lane * 4);
  float4 vd = *(const float4*)(h + (size_t)d * CCH + lane * 4);
  float* ad = agg + (size_t)d * CCH + lane * 4;
  float* as = agg + (size_t)s * CCH + lane * 4;
  atomicAdd(ad + 0, vs.x * coef); atomicAdd(ad + 1, vs.y * coef);
  atomicAdd(ad + 2, vs.z * coef); atomicAdd(ad + 3, vs.w * coef);
  atomicAdd(as + 0, vd.x * coef); atomicAdd(as + 1, vd.y * coef);
  atomicAdd(as + 2, vd.z * coef); atomicAdd(as + 3, vd.w * coef);
}

// ---------------- agg += h/deg + conv_b (in place), fused BN1 sum/sumsq ----------------
__global__ void __launch_bounds__(128) combine_stats_kernel(float* __restrict__ agg,
                                                            const float* __restrict__ h,
                                                            const float* __restrict__ deg,
                                                            const float* __restrict__ conv_b,
                                                            float* __restrict__ s1,
                                                            float* __restrict__ s2,
                                                            int rows_per_block, int n_rows) {
  int c  = threadIdx.x;
  int r0 = blockIdx.x * rows_per_block;
  int r1 = r0 + rows_per_block; if (r1 > n_rows) r1 = n_rows;
  float bias = conv_b[c];
  float sum = 0.f, sq = 0.f;
  for (int r = r0; r < r1; ++r) {
    float inv = 1.0f / deg[r];
    size_t i = (size_t)r * CCH + c;
    float v = agg[i] + h[i] * inv + bias;
    agg[i] = v;
    sum += v; sq += v * v;
  }
  atomicAdd(&s1[c], sum);
  atomicAdd(&s2[c], sq);
}

__global__ void bn_finalize_kernel(const float* __restrict__ s1, const float* __restrict__ s2,
                                   float n, float* __restrict__ mean, float* __restrict__ rstd) {
  int c = threadIdx.x;
  float m = s1[c] / n;
  float v = s2[c] / n - m * m;
  mean[c] = m;
  rstd[c] = rsqrtf(v + BN_EPS_C);
}

// ---------------- BN1-apply + segment max pool (float atomic max) ----------------
__global__ void pool_kernel(const float* __restrict__ agg, const int* __restrict__ batch,
                            const float* __restrict__ mean, const float* __restrict__ rstd,
                            const float* __restrict__ g, const float* __restrict__ b,
                            float* __restrict__ p, int n_rows) {
  long long i = (long long)blockIdx.x * blockDim.x + threadIdx.x;
  if (i >= (long long)n_rows * CCH) return;
  int n = (int)(i >> 7);
  int c = (int)(i & (CCH - 1));
  float v = (agg[i] - mean[c]) * rstd[c] * g[c] + b[c];
  atomicMax(&p[(size_t)batch[n] * CCH + c], v);
}

// ---------------- relu(p) + BN2 stats (single block, 128 threads) ----------------
__global__ void __launch_bounds__(128) relu_stats_kernel(const float* __restrict__ p,
                                                         float* __restrict__ prelu,
                                                         float* __restrict__ mean,
                                                         float* __restrict__ rstd) {
  int c = threadIdx.x;
  float sum = 0.f, sq = 0.f;
  for (int g = 0; g < NUM_GRAPHS_C; ++g) {
    float v = p[g * CCH + c];
    v = v > 0.f ? v : 0.f;
    prelu[g * CCH + c] = v;
    sum += v; sq += v * v;
  }
  float m   = sum / (float)NUM_GRAPHS_C;
  float var = sq / (float)NUM_GRAPHS_C - m * m;
  mean[c] = m;
  rstd[c] = rsqrtf(var + BN_EPS_C);
}

// ---------------- h2 = relu( BN2(prelu) @ fc_w + fc_b )  (chained WMMA, K=128) ----------------
__global__ void __launch_bounds__(32) fc_wmma_kernel(const float* __restrict__ a_in,
                                                     const float* __restrict__ mean,
                                                     const float* __restrict__ rstd,
                                                     const float* __restrict__ gam,
                                                     const float* __restrict__ bet,
                                                     const float* __restrict__ w,
                                                     const float* __restrict__ bias,
                                                     float* __restrict__ h2) {
  const int rt   = blockIdx.x >> 3;   // 32 row tiles of 512
  const int ct   = blockIdx.x & 7;    // 8 col tiles of 128
  const int lane = threadIdx.x;
  const int half = lane >> 4;
  const int m    = lane & 15;
  const int row  = rt * 16 + m;
  const int col  = ct * 16 + m;

  v8f acc = {};
  for (int kb = 0; kb < CCH / 4; ++kb) {
    const int k0 = kb * 4 + half * 2;
    v2f a, b;
    float x0 = a_in[row * CCH + k0];
    float x1 = a_in[row * CCH + k0 + 1];
    a.x = (x0 - mean[k0])     * rstd[k0]     * gam[k0]     + bet[k0];
    a.y = (x1 - mean[k0 + 1]) * rstd[k0 + 1] * gam[k0 + 1] + bet[k0 + 1];
    b.x = w[(k0)     * CCH + col];
    b.y = w[(k0 + 1) * CCH + col];
    acc = __builtin_amdgcn_wmma_f32_16x16x4_f32(false, a, false, b, (short)0, acc, false, false);
  }
  const int rbase = rt * 16 + half * 8;
#pragma unroll
  for (int r = 0; r < 8; ++r) {
    float v = acc[r] + bias[col];
    h2[(rbase + r) * CCH + col] = v > 0.f ? v : 0.f;
  }
}

// ---------------- BN3 stats (single block) ----------------
__global__ void __launch_bounds__(128) stats_kernel(const float* __restrict__ h2,
                                                    float* __restrict__ mean,
                                                    float* __restrict__ rstd) {
  int c = threadIdx.x;
  float sum = 0.f, sq = 0.f;
  for (int g = 0; g < NUM_GRAPHS_C; ++g) {
    float v = h2[g * CCH + c];
    sum += v; sq += v * v;
  }
  float m   = sum / (float)NUM_GRAPHS_C;
  float var = sq / (float)NUM_GRAPHS_C - m * m;
  mean[c] = m;
  rstd[c] = rsqrtf(var + BN_EPS_C);
}

// ---------------- o = BN3(h2) @ out_w + out_b ----------------
__global__ void out_kernel(const float* __restrict__ h2, const float* __restrict__ mean,
                           const float* __restrict__ rstd, const float* __restrict__ gam,
                           const float* __restrict__ bet, const float* __restrict__ ow,
                           const float* __restrict__ ob, float* __restrict__ out) {
  int g = blockIdx.x * blockDim.x + threadIdx.x;
  if (g >= NUM_GRAPHS_C) return;
  float acc = 0.f;
  for (int c = 0; c < CCH; ++c) {
    float v = (h2[g * CCH + c] - mean[c]) * rstd[c] * gam[c] + bet[c];
    acc += v * ow[c];
  }
  out[g] = acc + ob[0];
}

extern "C" void kernel_launch(void* const* d_in, const int* in_sizes, int n_in,
                              void* d_out, int out_size, void* d_ws, size_t ws_size,
                              hipStream_t stream) {
  (void)n_in; (void)out_size; (void)ws_size;
  const float* x      = (const float*)d_in[0];
  const int*   ei     = (const int*)d_in[1];
  const int*   batch  = (const int*)d_in[2];
  const float* conv_w = (const float*)d_in[3];
  const float* conv_b = (const float*)d_in[4];
  const float* bn1_g  = (const float*)d_in[5];
  const float* bn1_b  = (const float*)d_in[6];
  const float* fbn_g  = (const float*)d_in[7];
  const float* fbn_b  = (const float*)d_in[8];
  const float* fc_w   = (const float*)d_in[9];
  const float* fc_b   = (const float*)d_in[10];
  const float* obn_g  = (const float*)d_in[11];
  const float* obn_b  = (const float*)d_in[12];
  const float* out_w  = (const float*)d_in[13];
  const float* out_b  = (const float*)d_in[14];
  float* out = (float*)d_out;

  const int N = in_sizes[2];          // 200000 nodes
  const int E = in_sizes[1] / 2;      // 1.6M directed edges
  const int*  src = ei;
  const int*  dst = ei + E;

  // workspace layout (floats)
  float* ws    = (float*)d_ws;
  float* h     = ws;                  ws += (size_t)N * CCH;       // 25.6M
  float* agg   = ws;                  ws += (size_t)N * CCH;       // 25.6M (becomes BN1 input in place)
  float* deg   = ws;                  ws += N;
  float* dinv  = ws;                  ws += N;
  float* s1    = ws;                  ws += CCH;
  float* s2    = ws;                  ws += CCH;
  float* mean1 = ws;                  ws += CCH;
  float* rstd1 = ws;                  ws += CCH;
  float* p     = ws;                  ws += NUM_GRAPHS_C * CCH;
  float* prelu = ws;                  ws += NUM_GRAPHS_C * CCH;
  float* mean2 = ws;                  ws += CCH;
  float* rstd2 = ws;                  ws += CCH;
  float* h2    = ws;                  ws += NUM_GRAPHS_C * CCH;
  float* mean3 = ws;                  ws += CCH;
  float* rstd3 = ws;                  ws += CCH;

  // ---- init (must re-zero every call: atomics accumulate) ----
  {
    long long nAgg = (long long)N * CCH;
    fill_f32<<<(int)((nAgg + 255) / 256), 256, 0, stream>>>(agg, 0.0f, (int)nAgg);
    fill_f32<<<(N + 255) / 256, 256, 0, stream>>>(deg, 1.0f, N);               // self-loop
    fill_f32<<<1, 256, 0, stream>>>(s1, 0.0f, 2 * CCH);                        // s1|s2 contiguous
    fill_f32<<<(NUM_GRAPHS_C * CCH + 255) / 256, 256, 0, stream>>>(p, -INFINITY, NUM_GRAPHS_C * CCH);
  }

  // ---- h = x @ conv_w (WMMA) ----
  transform_wmma<<<N / 16, 32, 0, stream>>>(x, conv_w, h);

  // ---- degrees / dinv ----
  degree_kernel<<<(E + 255) / 256, 256, 0, stream>>>(src, dst, deg, E);
  dinv_kernel<<<(N + 255) / 256, 256, 0, stream>>>(deg, dinv, N);

  // ---- edge scatter (dominant, L2-resident atomics) ----
  scatter_kernel<<<(E * 32 + 255) / 256, 256, 0, stream>>>(src, dst, h, dinv, agg, E);

  // ---- self-term + bias + BN1 stats ----
  const int RPB = 64;
  combine_stats_kernel<<<(N + RPB - 1) / RPB, 128, 0, stream>>>(agg, h, deg, conv_b, s1, s2, RPB, N);
  bn_finalize_kernel<<<1, 128, 0, stream>>>(s1, s2, (float)N, mean1, rstd1);

  // ---- BN1-apply + global max pool ----
  {
    long long nEl = (long long)N * CCH;
    pool_kernel<<<(int)((nEl + 255) / 256), 256, 0, stream>>>(agg, batch, mean1, rstd1, bn1_g, bn1_b, p, N);
  }

  // ---- head: relu -> BN2 -> fc(WMMA)+relu -> BN3 -> out ----
  relu_stats_kernel<<<1, 128, 0, stream>>>(p, prelu, mean2, rstd2);
  fc_wmma_kernel<<<(NUM_GRAPHS_C / 16) * (CCH / 16), 32, 0, stream>>>(prelu, mean2, rstd2,
                                                                      fbn_g, fbn_b, fc_w, fc_b, h2);
  stats_kernel<<<1, 128, 0, stream>>>(h2, mean3, rstd3);
  out_kernel<<<(NUM_GRAPHS_C + 127) / 128, 128, 0, stream>>>(h2, mean3, rstd3, obn_g, obn_b,
                                                             out_w, out_b, out);
}